// DaviesBouldinLossFunction_64269890617637
// MI455X (gfx1250) — compile-verified
//
#include <hip/hip_runtime.h>
#include <cstdint>
#include <cstddef>

#define CLS   64
#define DIM   256
#define NROW  262144
#define MAXG  512
#define XSTR  260   // padded LDS stride for kernel 3 (bank-conflict dodge)

typedef __attribute__((ext_vector_type(2))) float v2f;
typedef __attribute__((ext_vector_type(8))) float v8f;

// ---------------------------------------------------------------------------
// Kernel 1: streaming pass over predicted[N][D].
// Block = 256 threads; thread d owns column d of a [CLS][DIM] LDS accumulator
// (exclusive ownership -> deterministic, atomic-free scatter-add).
// Per row: pr = predicted/count[t]; acc[t][d] += pr; vec = ||centroid[t]-pr||
// accumulated into svec[t] by thread 0 (fixed row order -> deterministic).
// 4 rows per iteration to amortize barriers. Partials written per block.
// ---------------------------------------------------------------------------
__global__ __launch_bounds__(256) void k1_accumulate(
    const float* __restrict__ predicted,
    const float* __restrict__ centroids,
    const float* __restrict__ count,
    const int*   __restrict__ target,
    float* __restrict__ partial_acc,   // [G][CLS*DIM]
    float* __restrict__ partial_s,     // [G][CLS]
    int rowsPerBlock)
{
    __shared__ float acc[CLS * DIM];   // 64 KB
    __shared__ float svec[CLS];
    __shared__ float invcnt[CLS];
    __shared__ float red[4 * 8];

    const int tid  = threadIdx.x;
    const int blk  = blockIdx.x;
    const int wid  = tid >> 5;
    const int lane = tid & 31;

    for (int c = 0; c < CLS; ++c) acc[c * DIM + tid] = 0.0f;
    if (tid < CLS) { svec[tid] = 0.0f; invcnt[tid] = 1.0f / count[tid]; }
    __syncthreads();

    int rowStart = blk * rowsPerBlock;
    int rowEnd   = rowStart + rowsPerBlock;
    if (rowEnd > NROW) rowEnd = NROW;

    for (int r = rowStart; r < rowEnd; r += 4) {
        float sq[4];
#pragma unroll
        for (int j = 0; j < 4; ++j) {
            int row = r + j;                 // uniform across block
            float sqv = 0.0f;
            if (row < rowEnd) {
                int   t  = target[row];      // broadcast load
                float p  = predicted[(size_t)row * DIM + tid];
                float pr = p * invcnt[t];
                acc[t * DIM + tid] += pr;    // exclusive per-thread column
                float diff = centroids[t * DIM + tid] - pr;
                sqv = diff * diff;
            }
            sq[j] = sqv;
        }
#pragma unroll
        for (int j = 0; j < 4; ++j) {
            float v = sq[j];
            for (int off = 16; off > 0; off >>= 1)
                v += __shfl_xor(v, off, 32);
            if (lane == 0) red[j * 8 + wid] = v;
        }
        __syncthreads();
        if (tid == 0) {
#pragma unroll
            for (int j = 0; j < 4; ++j) {
                int row = r + j;
                if (row < rowEnd) {
                    float tot = 0.0f;
                    for (int w = 0; w < 8; ++w) tot += red[j * 8 + w];
                    svec[target[row]] += sqrtf(tot);
                }
            }
        }
        __syncthreads();
    }

    for (int c = 0; c < CLS; ++c)
        partial_acc[(size_t)blk * (CLS * DIM) + c * DIM + tid] = acc[c * DIM + tid];
    if (tid < CLS)
        partial_s[blk * CLS + tid] = svec[tid];
}

// ---------------------------------------------------------------------------
// Kernel 2: 64 blocks (one per class), 256 threads (one per column).
// Reduce partials in fixed g-order -> cent_new; per-class sum of squares,
// sum of |.|, and s = sqrt(distances + sum vec) / count.
// ---------------------------------------------------------------------------
__global__ __launch_bounds__(256) void k2_finalize_classes(
    const float* __restrict__ centroids,
    const float* __restrict__ distances,
    const float* __restrict__ count,
    const float* __restrict__ partial_acc,
    const float* __restrict__ partial_s,
    float* __restrict__ cent_new,   // [CLS*DIM]
    float* __restrict__ sq_out,     // [CLS]
    float* __restrict__ abs_out,    // [CLS]
    float* __restrict__ s_out,      // [CLS]
    int G)
{
    __shared__ float redA[8];
    __shared__ float redS[8];
    const int c    = blockIdx.x;
    const int d    = threadIdx.x;
    const int wid  = d >> 5;
    const int lane = d & 31;

    float sum = centroids[c * DIM + d];
    for (int g = 0; g < G; ++g)
        sum += partial_acc[(size_t)g * (CLS * DIM) + c * DIM + d];
    cent_new[c * DIM + d] = sum;

    float a  = fabsf(sum);
    float s2 = sum * sum;
    for (int off = 16; off > 0; off >>= 1) {
        a  += __shfl_xor(a,  off, 32);
        s2 += __shfl_xor(s2, off, 32);
    }
    if (lane == 0) { redA[wid] = a; redS[wid] = s2; }
    __syncthreads();
    if (d == 0) {
        float at = 0.0f, st = 0.0f;
        for (int w = 0; w < 8; ++w) { at += redA[w]; st += redS[w]; }
        abs_out[c] = at;
        sq_out[c]  = st;
        float sacc = distances[c];
        for (int g = 0; g < G; ++g) sacc += partial_s[g * CLS + c];
        s_out[c] = sqrtf(sacc) / count[c];
    }
}

// ---------------------------------------------------------------------------
// Kernel 3: single block, 512 threads = 16 waves. Each wave computes one
// 16x16 tile of gram = cent_new @ cent_new^T (K=256) with
// V_WMMA_F32_16X16X4_F32 (64 WMMAs per wave), then applies the Davies-
// Bouldin masked term and reduces to the scalar loss.
// A 16x4 layout: lane(l<16): M=l, VGPR v -> K = v + 2*(l>=16).
// B 4x16 layout: lane gives N, VGPR v -> K = v + 2*(l>=16).  (ISA 7.12.2)
// C/D: VGPR v, lanes 0-15 -> M=v, lanes 16-31 -> M=v+8; N=lane&15.
// ---------------------------------------------------------------------------
__global__ __launch_bounds__(512) void k3_loss(
    const float* __restrict__ cent_new,
    const float* __restrict__ sq_in,
    const float* __restrict__ abs_in,
    const float* __restrict__ s_in,
    const float* __restrict__ class_weights,
    float* __restrict__ out)
{
    __shared__ float Xl[CLS * XSTR];   // padded stride
    __shared__ float sl[CLS];
    __shared__ float sql[CLS];
    __shared__ float red[16];

    const int tid = threadIdx.x;
    for (int idx = tid; idx < CLS * DIM; idx += 512) {
        int row = idx >> 8;
        int col = idx & 255;
        Xl[row * XSTR + col] = cent_new[idx];
    }
    if (tid < CLS) { sl[tid] = s_in[tid]; sql[tid] = sq_in[tid]; }
    __syncthreads();

    const int wid  = tid >> 5;
    const int lane = tid & 31;
    const int ti   = wid >> 2;       // tile row  (0..3)
    const int tj   = wid & 3;        // tile col  (0..3)
    const int half = lane >> 4;      // 0: lanes 0-15, 1: lanes 16-31
    const int l15  = lane & 15;
    const int rowA = ti * 16 + l15;  // gram row source
    const int rowB = tj * 16 + l15;  // gram col source (B = X^T slice)

    v8f cacc = (v8f){0.f,0.f,0.f,0.f,0.f,0.f,0.f,0.f};
    for (int k0 = 0; k0 < DIM; k0 += 4) {
        int ka = k0 + half * 2;
        v2f a, b;
        a.x = Xl[rowA * XSTR + ka];
        a.y = Xl[rowA * XSTR + ka + 1];
        b.x = Xl[rowB * XSTR + ka];
        b.y = Xl[rowB * XSTR + ka + 1];
        cacc = __builtin_amdgcn_wmma_f32_16x16x4_f32(
            /*neg_a=*/false, a, /*neg_b=*/false, b,
            /*c_mod=*/(short)0, cacc, /*reuse_a=*/false, /*reuse_b=*/false);
    }

    float local = 0.0f;
#pragma unroll
    for (int v = 0; v < 8; ++v) {
        int i = ti * 16 + v + half * 8;
        int j = tj * 16 + l15;
        if (i != j) {
            float d2 = sql[i] + sql[j] - 2.0f * cacc[v];
            d2 = fmaxf(d2, 0.0f);
            float m = sqrtf(d2);
            local += class_weights[i * CLS + j] * (sl[i] + sl[j]) / m;
        }
    }
    for (int off = 16; off > 0; off >>= 1)
        local += __shfl_xor(local, off, 32);
    if (lane == 0) red[wid] = local;
    __syncthreads();
    if (tid == 0) {
        float tot = 0.0f;
        for (int w = 0; w < 16; ++w) tot += red[w];
        float loss = tot / (float)CLS * (float)(CLS - 1);
        float abssum = 0.0f;
        for (int c = 0; c < CLS; ++c) abssum += abs_in[c];
        out[0] = loss + abssum * 1.0e-6f;
    }
}

// ---------------------------------------------------------------------------
extern "C" void kernel_launch(void* const* d_in, const int* in_sizes, int n_in,
                              void* d_out, int out_size, void* d_ws, size_t ws_size,
                              hipStream_t stream)
{
    (void)in_sizes; (void)n_in; (void)out_size;

    const float* predicted     = (const float*)d_in[0];
    const float* centroids     = (const float*)d_in[1];
    const float* distances     = (const float*)d_in[2];
    const float* count         = (const float*)d_in[3];
    const float* class_weights = (const float*)d_in[4];
    const int*   target        = (const int*)d_in[5];
    float*       out           = (float*)d_out;

    // Workspace layout (floats): [G*CLS*DIM partial_acc][G*CLS partial_s]
    //                            [CLS*DIM cent_new][CLS sq][CLS abs][CLS s]
    const size_t fixedFloats = (size_t)CLS * DIM + 3 * CLS;
    const size_t perG        = (size_t)CLS * DIM + CLS;
    const size_t availFloats = ws_size / sizeof(float);

    int G = MAXG;
    if (availFloats <= fixedFloats + perG) {
        G = 1;
    } else {
        size_t gmax = (availFloats - fixedFloats) / perG;
        if (gmax < (size_t)G) G = (int)gmax;
        if (G < 1) G = 1;
    }

    float* ws          = (float*)d_ws;
    float* partial_acc = ws;
    float* partial_s   = partial_acc + (size_t)G * CLS * DIM;
    float* cent_new    = partial_s   + (size_t)G * CLS;
    float* sq_buf      = cent_new    + (size_t)CLS * DIM;
    float* abs_buf     = sq_buf      + CLS;
    float* s_buf       = abs_buf     + CLS;

    int rpb = (NROW + G - 1) / G;
    rpb = (rpb + 3) & ~3;            // multiple of 4 (row batching)

    k1_accumulate<<<G, 256, 0, stream>>>(predicted, centroids, count, target,
                                         partial_acc, partial_s, rpb);
    k2_finalize_classes<<<CLS, 256, 0, stream>>>(centroids, distances, count,
                                                 partial_acc, partial_s,
                                                 cent_new, sq_buf, abs_buf, s_buf, G);
    k3_loss<<<1, 512, 0, stream>>>(cent_new, sq_buf, abs_buf, s_buf,
                                   class_weights, out);
}